// MPEmbedding_9405978378554
// MI455X (gfx1250) — compile-verified
//
#include <hip/hip_runtime.h>
#include <stdint.h>

// MP-embedding: out[r, :] = weight[idx[r], :] * rsqrt(mean(weight[idx[r], :]^2) + 1e-4)
// Memory-bound gather+normalize. CDNA5 path: async HBM->LDS row DMA
// (global_load_async_to_lds_b128, ASYNCcnt), double-buffered per wave,
// non-temporal b128 stores for the write-once output.

#define DIM             1024
#define ROW_BYTES       (DIM * 4)          // 4096 B per row
#define WAVES_PER_BLOCK 8
#define BLOCK_THREADS   (WAVES_PER_BLOCK * 32)
#define CHUNK_BYTES     512                // 32 lanes x 16 B per async b128 issue
#define CHUNKS_PER_ROW  (ROW_BYTES / CHUNK_BYTES)   // 8

typedef float v4f __attribute__((ext_vector_type(4)));

__global__ __launch_bounds__(BLOCK_THREADS)
void mp_embedding_kernel(const int* __restrict__ idx,
                         const float* __restrict__ weight,
                         float* __restrict__ out,
                         int nrows)
{
    // 2 row buffers per wave (double buffering), 4 KB each.
    __shared__ __align__(16) float lds[WAVES_PER_BLOCK * 2 * DIM];

    const int lane  = threadIdx.x & 31;
    const int wave  = threadIdx.x >> 5;
    const int gwave = blockIdx.x * WAVES_PER_BLOCK + wave;
    const int nwave = gridDim.x * WAVES_PER_BLOCK;

    float* mybuf = &lds[wave * 2 * DIM];
    // Low 32 bits of the flat address of an LDS object == workgroup-relative
    // LDS byte offset (flat-LDS aperture layout); that is exactly what the
    // async instruction's VDST (LDS address VGPR) wants.
    const uint32_t lds_base = (uint32_t)(uintptr_t)mybuf;
    const uint64_t wbase    = (uint64_t)(uintptr_t)weight;

    // Issue the 8 async b128 transfers that gather one table row into LDS buf.
    auto issue_row = [&](int row, int buf) {
        const uint32_t r    = (uint32_t)idx[row];            // wave-uniform value
        uint32_t       goff = r * (uint32_t)ROW_BYTES + (uint32_t)(lane << 4);
        uint32_t       loff = lds_base + (uint32_t)buf * ROW_BYTES + (uint32_t)(lane << 4);
#pragma unroll
        for (int c = 0; c < CHUNKS_PER_ROW; ++c) {
            asm volatile("global_load_async_to_lds_b128 %0, %1, %2"
                         :
                         : "v"(loff), "v"(goff), "s"(wbase)
                         : "memory");
            goff += CHUNK_BYTES;
            loff += CHUNK_BYTES;
        }
    };

    int row = gwave;
    if (row >= nrows) return;

    issue_row(row, 0);
    int buf = 0;

    for (;;) {
        const int next = row + nwave;
        if (next < nrows) {
            issue_row(next, buf ^ 1);
            // In-order completion: <=8 outstanding means the current row's
            // 8 transfers (issued earlier) have landed in LDS.
            asm volatile("s_wait_asynccnt 0x8" ::: "memory");
        } else {
            asm volatile("s_wait_asynccnt 0x0" ::: "memory");
        }

        // ---- compute: sum of squares over the row (lane-strided float4) ----
        const v4f* src = (const v4f*)(mybuf + buf * DIM);
        v4f   v[CHUNKS_PER_ROW];
        float ss = 0.0f;
#pragma unroll
        for (int c = 0; c < CHUNKS_PER_ROW; ++c) {
            v[c] = src[c * 32 + lane];                       // ds_load_b128
            ss += v[c].x * v[c].x + v[c].y * v[c].y
                + v[c].z * v[c].z + v[c].w * v[c].w;
        }
        // wave32 butterfly reduction
#pragma unroll
        for (int m = 16; m > 0; m >>= 1)
            ss += __shfl_xor(ss, m, 32);

        const float inv = 1.0f / sqrtf(ss * (1.0f / (float)DIM) + 1e-4f);

        // ---- scale + non-temporal b128 stores (output is write-once) ----
        v4f* dst = (v4f*)(out + (size_t)row * DIM);
#pragma unroll
        for (int c = 0; c < CHUNKS_PER_ROW; ++c) {
            v4f o = v[c] * inv;
            __builtin_nontemporal_store(o, &dst[c * 32 + lane]);
        }

        if (next >= nrows) break;
        row = next;
        buf ^= 1;
    }
}

extern "C" void kernel_launch(void* const* d_in, const int* in_sizes, int n_in,
                              void* d_out, int out_size, void* d_ws, size_t ws_size,
                              hipStream_t stream) {
    const int*   x      = (const int*)d_in[0];     // [batch*seq] indices
    const float* weight = (const float*)d_in[1];   // [50257, 1024] fp32 table
    float*       out    = (float*)d_out;           // [batch*seq, 1024] fp32
    const int    nrows  = in_sizes[0];

    int grid = (nrows + WAVES_PER_BLOCK - 1) / WAVES_PER_BLOCK;
    if (grid > 1024) grid = 1024;   // 8192 waves -> 2 rows/wave, pipelined
    if (grid < 1)    grid = 1;

    mp_embedding_kernel<<<dim3(grid), dim3(BLOCK_THREADS), 0, stream>>>(
        x, weight, out, nrows);
}